// RoutingCapsules_10505490006698
// MI455X (gfx1250) — compile-verified
//
#include <hip/hip_runtime.h>
#include <math.h>

typedef __attribute__((ext_vector_type(2))) float v2f;
typedef __attribute__((ext_vector_type(8))) float v8f;

#define BATCH 256
#define NCAPS 10
#define ICAPS 1152
#define DCAPS 16
#define KDIM  8

// ---------------------------------------------------------------------------
// Kernel A: u_hat[b,n,i,d] = sum_k x[b,i,k] * W[n,i,d,k]
// Per (n,i, 16-batch tile): C[16b x 16d] = A[16b x 8k] * B[8k x 16d]
// via two V_WMMA_F32_16X16X4_F32 (K=0..3 and K=4..7).
// Grid: 16 btiles * 10 n * 16 chunks = 2560 blocks of 256 threads (8 waves),
// each wave handles 9 consecutive i values (2 WMMAs per i).
// ---------------------------------------------------------------------------
__global__ __launch_bounds__(256) void uhat_wmma_kernel(
    const float* __restrict__ x,   // [256][1152][8]
    const float* __restrict__ W,   // [10][1152][16][8]
    float* __restrict__ u)         // [256][10][1152][16]
{
    const int lane = threadIdx.x & 31;
    const int wave = threadIdx.x >> 5;
    const int row  = lane & 15;     // M (batch) for A, N (dim_caps) for B/C
    const int hi   = lane >> 4;     // 0 -> K={0,1}; 1 -> K={2,3} (per f32 WMMA layout)

    int blk = blockIdx.x;
    const int chunk = blk & 15;  blk >>= 4;
    const int n  = blk % NCAPS;
    const int bt = blk / NCAPS;
    const int i0 = (chunk * 8 + wave) * 9;

    const int bA = bt * 16 + row;                       // this lane's A-row batch
    const float* xrow = x + (size_t)bA * (ICAPS * KDIM);
    const float* wrow = W + (size_t)n * (ICAPS * DCAPS * KDIM) + (size_t)row * KDIM;
    const int k0 = 2 * hi;

    for (int ii = 0; ii < 9; ++ii) {
        const int i = i0 + ii;
        const float* xp = xrow + (size_t)i * KDIM;
        const float* wp = wrow + (size_t)i * (DCAPS * KDIM);

        // A: 16x4 f32, lane holds A[M=row][K=k0], A[M=row][K=k0+1]
        v2f a0 = { xp[k0],     xp[k0 + 1] };
        v2f a1 = { xp[k0 + 4], xp[k0 + 5] };
        // B: 4x16 f32, lane holds B[K=k0][N=row] = W[n,i,row,k0], etc.
        v2f b0 = { wp[k0],     wp[k0 + 1] };
        v2f b1 = { wp[k0 + 4], wp[k0 + 5] };

        v8f c = {0.f, 0.f, 0.f, 0.f, 0.f, 0.f, 0.f, 0.f};
        c = __builtin_amdgcn_wmma_f32_16x16x4_f32(false, a0, false, b0, (short)0, c, false, false);
        c = __builtin_amdgcn_wmma_f32_16x16x4_f32(false, a1, false, b1, (short)0, c, false, false);

        // C/D layout: VGPR r -> M = r (lanes 0-15), M = r+8 (lanes 16-31); N = row
#pragma unroll
        for (int r = 0; r < 8; ++r) {
            const int bo = bt * 16 + r + 8 * hi;
            u[(((size_t)bo * NCAPS + n) * ICAPS + i) * DCAPS + row] = c[r];
        }
    }
}

// ---------------------------------------------------------------------------
// c[b,n,i] = softmax over n of blog[b,n,i]   (n stride = ICAPS)
// One thread per (b,i): 294912 threads.
// ---------------------------------------------------------------------------
__global__ __launch_bounds__(256) void softmax_n_kernel(
    const float* __restrict__ blog, float* __restrict__ c)
{
    const int idx = blockIdx.x * 256 + threadIdx.x;       // b*ICAPS + i
    const int b = idx / ICAPS;
    const int i = idx % ICAPS;
    const float* lp = blog + (size_t)b * NCAPS * ICAPS + i;

    float l[NCAPS];
    float m = -INFINITY;
#pragma unroll
    for (int n = 0; n < NCAPS; ++n) {
        l[n] = lp[(size_t)n * ICAPS];
        m = fmaxf(m, l[n]);
    }
    float s = 0.f;
#pragma unroll
    for (int n = 0; n < NCAPS; ++n) { l[n] = expf(l[n] - m); s += l[n]; }
    const float inv = 1.f / s;
    float* cp = c + (size_t)b * NCAPS * ICAPS + i;
#pragma unroll
    for (int n = 0; n < NCAPS; ++n) cp[(size_t)n * ICAPS] = l[n] * inv;
}

// ---------------------------------------------------------------------------
// s[b,n,d] = sum_i c[b,n,i]*u[b,n,i,d]; out[b,n,d] = squash(s) over d.
// One block of 128 threads per (b,n): 2560 blocks. Thread = (group g=tid/16, d=tid%16).
// ---------------------------------------------------------------------------
__global__ __launch_bounds__(128) void s_squash_kernel(
    const float* __restrict__ cvec, const float* __restrict__ u,
    float* __restrict__ out)
{
    __shared__ float red[128];
    const int bn = blockIdx.x;                  // b*NCAPS + n
    const int d = threadIdx.x & 15;
    const int g = threadIdx.x >> 4;             // 8 groups
    const float* up = u    + (size_t)bn * ICAPS * DCAPS;
    const float* cp = cvec + (size_t)bn * ICAPS;

    float acc = 0.f;
    for (int i = g; i < ICAPS; i += 8)
        acc += cp[i] * up[(size_t)i * DCAPS + d];
    red[threadIdx.x] = acc;
    __syncthreads();

    if (threadIdx.x < 16) {
        float s = 0.f;
#pragma unroll
        for (int r = 0; r < 8; ++r) s += red[r * 16 + threadIdx.x];
        red[threadIdx.x] = s;                   // s[d] now in red[0..15]
    }
    __syncthreads();

    if (threadIdx.x < 16) {
        const float s = red[threadIdx.x];
        float sq = 0.f;
#pragma unroll
        for (int j = 0; j < 16; ++j) sq += red[j] * red[j];
        const float norm = sqrtf(sq);
        out[(size_t)bn * DCAPS + threadIdx.x] = (sq / (1.f + sq)) * s / (norm + 1e-8f);
    }
}

// ---------------------------------------------------------------------------
// blog[b,n,i] += sum_d u[b,n,i,d] * v[b,n,d].  One thread per (b,n,i).
// ---------------------------------------------------------------------------
__global__ __launch_bounds__(256) void bupdate_kernel(
    const float* __restrict__ u, const float* __restrict__ v,
    float* __restrict__ blog)
{
    const size_t idx = (size_t)blockIdx.x * 256 + threadIdx.x;  // (b*10+n)*ICAPS + i
    const size_t bn = idx / ICAPS;
    const float4* up = (const float4*)(u + idx * DCAPS);
    const float4* vp = (const float4*)(v + bn * DCAPS);
    float dot = 0.f;
#pragma unroll
    for (int q = 0; q < 4; ++q) {
        const float4 a = up[q], b = vp[q];
        dot += a.x * b.x + a.y * b.y + a.z * b.z + a.w * b.w;
    }
    blog[idx] += dot;
}

// ---------------------------------------------------------------------------
extern "C" void kernel_launch(void* const* d_in, const int* in_sizes, int n_in,
                              void* d_out, int out_size, void* d_ws, size_t ws_size,
                              hipStream_t stream) {
    const float* x = (const float*)d_in[0];   // [256,1152,8]
    const float* W = (const float*)d_in[1];   // [1,10,1152,16,8]
    float* out = (float*)d_out;               // [256,10,16]

    float* ws   = (float*)d_ws;
    float* u    = ws;                                              // 47,185,920 f
    float* blog = u    + (size_t)BATCH * NCAPS * ICAPS * DCAPS;    //  2,949,120 f
    float* cbuf = blog + (size_t)BATCH * NCAPS * ICAPS;            //  2,949,120 f
    float* v    = cbuf + (size_t)BATCH * NCAPS * ICAPS;            //     40,960 f

    hipMemsetAsync(blog, 0, (size_t)BATCH * NCAPS * ICAPS * sizeof(float), stream);

    // u_hat via fp32 WMMA (16 btiles * 10 caps * 16 chunks)
    uhat_wmma_kernel<<<16 * NCAPS * 16, 256, 0, stream>>>(x, W, u);

    const int sm_blocks = BATCH * ICAPS / 256;   // 1152
    const int bn_blocks = BATCH * NCAPS;         // 2560
    const int bu_blocks = BATCH * NCAPS * ICAPS / 256; // 11520

    for (int t = 0; t < 3; ++t) {
        softmax_n_kernel<<<sm_blocks, 256, 0, stream>>>(blog, cbuf);
        s_squash_kernel<<<bn_blocks, 128, 0, stream>>>(cbuf, u, v);
        bupdate_kernel<<<bu_blocks, 256, 0, stream>>>(u, v, blog);
    }
    softmax_n_kernel<<<sm_blocks, 256, 0, stream>>>(blog, cbuf);
    s_squash_kernel<<<bn_blocks, 128, 0, stream>>>(cbuf, u, out);
}